// PairLoss_15702400434564
// MI455X (gfx1250) — compile-verified
//
#include <hip/hip_runtime.h>

typedef __attribute__((ext_vector_type(2))) float v2f;
typedef __attribute__((ext_vector_type(8))) float v8f;

#define N 8192
#define M 128
#define KLBL 16
#define MARGIN_F 1.5f
#define INV_SQRT_M 0.08838834764831845f   // 1/sqrt(128)

#define BM 64
#define BN 32
#define LDSS 132   // 128 + 4 floats pad -> conflict-free, 528B row (16B aligned)

// ---------------------------------------------------------------------------
// Kernel 0: zero the scalar output (harness poisons d_out; we accumulate).
// ---------------------------------------------------------------------------
__global__ void zero_out_kernel(float* out) { out[0] = 0.0f; }

// ---------------------------------------------------------------------------
// Kernel 1: per-row squared norm + packed 16-bit label mask.
// One wave32 per row: 32 lanes x float4 = 128 floats.
// ---------------------------------------------------------------------------
__global__ void __launch_bounds__(256)
prep_kernel(const float* __restrict__ feature, const int* __restrict__ label,
            float* __restrict__ sq, unsigned* __restrict__ mask) {
  int wave = (blockIdx.x * blockDim.x + threadIdx.x) >> 5;
  int lane = threadIdx.x & 31;
  if (wave >= N) return;
  const float4* f4 = (const float4*)(feature + (size_t)wave * M);
  float4 v = f4[lane];
  float s = v.x * v.x + v.y * v.y + v.z * v.z + v.w * v.w;
#pragma unroll
  for (int o = 16; o > 0; o >>= 1) s += __shfl_xor(s, o, 32);
  if (lane == 0) {
    unsigned m = 0;
    const int* lb = label + (size_t)wave * KLBL;
#pragma unroll
    for (int t = 0; t < KLBL; ++t)
      if (lb[t] > 0) m |= (1u << t);
    sq[wave] = s;
    mask[wave] = m;
  }
}

// ---------------------------------------------------------------------------
// Kernel 2: tiled Gram via V_WMMA_F32_16X16X4_F32 + fused epilogue + reduce.
// Block = 256 threads = 8 waves, tile 64(M) x 32(N), one 16x16 per wave.
// ---------------------------------------------------------------------------
__global__ void __launch_bounds__(256)
pairloss_kernel(const float* __restrict__ feature, const float* __restrict__ sq,
                const unsigned* __restrict__ mask, float* __restrict__ out) {
  __shared__ float lsA[BM * LDSS];
  __shared__ float lsB[BN * LDSS];
  __shared__ float sqA[BM];
  __shared__ float sqB[BN];
  __shared__ unsigned mAa[BM];
  __shared__ unsigned mBa[BN];
  __shared__ float redbuf[8];

  const int tid = threadIdx.x;
  const int i0 = blockIdx.x * BM;
  const int j0 = blockIdx.y * BN;

  // ---- stage feature panels into LDS (float4, 16B-aligned since 528%16==0)
  {
    const float4* srcA = (const float4*)(feature + (size_t)i0 * M);
#pragma unroll
    for (int it = 0; it < 8; ++it) {
      int idx = tid + it * 256;          // float4 index; 32 float4 per row
      int row = idx >> 5;
      int c4 = idx & 31;
      float4 v = srcA[row * 32 + c4];
      *(float4*)&lsA[row * LDSS + c4 * 4] = v;
    }
    const float4* srcB = (const float4*)(feature + (size_t)j0 * M);
#pragma unroll
    for (int it = 0; it < 4; ++it) {
      int idx = tid + it * 256;
      int row = idx >> 5;
      int c4 = idx & 31;
      float4 v = srcB[row * 32 + c4];
      *(float4*)&lsB[row * LDSS + c4 * 4] = v;
    }
    if (tid < BM) {
      sqA[tid] = sq[i0 + tid];
      mAa[tid] = mask[i0 + tid];
    } else if (tid < BM + BN) {
      int t = tid - BM;
      sqB[t] = sq[j0 + t];
      mBa[t] = mask[j0 + t];
    }
  }
  __syncthreads();

  const int wave = tid >> 5;
  const int lane = tid & 31;
  const int wr = wave & 3;   // which 16-row subtile (0..3)
  const int wc = wave >> 2;  // which 16-col subtile (0..1)
  const int hi = lane >> 4;  // lanes 16-31 handle K+2 / M+8 halves
  const int l15 = lane & 15;

  // f32 WMMA fragment addressing (ISA 7.12.2, 32-bit A 16x4):
  //   lane -> row = lane&15 ; float2 at K = k + 2*(lane>>4)
  // B[k][n] = F[j0+n][k] gives the identical pattern on the B panel.
  const float* aBase = &lsA[(wr * 16 + l15) * LDSS + 2 * hi];
  const float* bBase = &lsB[(wc * 16 + l15) * LDSS + 2 * hi];

  v8f acc = {};
#pragma unroll
  for (int k = 0; k < M; k += 4) {
    v2f a = *(const v2f*)(aBase + k);
    v2f b = *(const v2f*)(bBase + k);
    acc = __builtin_amdgcn_wmma_f32_16x16x4_f32(
        /*neg_a=*/false, a, /*neg_b=*/false, b,
        /*c_mod=*/(short)0, acc, /*reuse_a=*/false, /*reuse_b=*/false);
  }

  // ---- fused epilogue: C/D layout => VGPR g, lane: M = g + 8*hi, N = lane&15
  const int jl = wc * 16 + l15;
  const float sqj = sqB[jl];
  const unsigned mj = mBa[jl];
  const int npj = __popc(mj);

  float lsum = 0.0f;
#pragma unroll
  for (int g = 0; g < 8; ++g) {
    int il = wr * 16 + g + 8 * hi;
    float sqi = sqA[il];
    unsigned mi = mAa[il];
    float inter = (float)__popc(mi & mj);
    float uni = (float)(__popc(mi) + npj) - inter;
    float w = inter / uni;            // 0/0 -> NaN (matches reference)
    w = (w == 0.0f) ? -1.0f : w;      // weight==0 -> -1 (NaN passes through)
    float d2 = sqi + sqj - 2.0f * acc[g];
    d2 = fmaxf(d2, 0.0f);
    float dist = (d2 > 0.0f) ? sqrtf(d2) : 0.0f;
    dist *= INV_SQRT_M;
    lsum += w * fminf(dist, MARGIN_F);
  }

  // ---- block reduction + single atomic per block
#pragma unroll
  for (int o = 16; o > 0; o >>= 1) lsum += __shfl_xor(lsum, o, 32);
  if (lane == 0) redbuf[wave] = lsum;
  __syncthreads();
  if (tid == 0) {
    float t = 0.0f;
#pragma unroll
    for (int wv = 0; wv < 8; ++wv) t += redbuf[wv];
    atomicAdd(out, t * (1.0f / ((float)N * (float)N)));
  }
}

// ---------------------------------------------------------------------------
extern "C" void kernel_launch(void* const* d_in, const int* in_sizes, int n_in,
                              void* d_out, int out_size, void* d_ws, size_t ws_size,
                              hipStream_t stream) {
  const float* feature = (const float*)d_in[0];  // [N, M] f32
  const int* label = (const int*)d_in[1];        // [N, K] i32
  float* out = (float*)d_out;                    // scalar f32

  float* sq = (float*)d_ws;                 // N floats
  unsigned* mask = (unsigned*)((char*)d_ws + (size_t)N * sizeof(float));  // N u32

  zero_out_kernel<<<1, 1, 0, stream>>>(out);

  // one wave per row: 8192 waves / 8 waves-per-block = 1024 blocks
  prep_kernel<<<N / 8, 256, 0, stream>>>(feature, label, sq, mask);

  dim3 grid(N / BM, N / BN);  // 128 x 256 blocks
  pairloss_kernel<<<grid, 256, 0, stream>>>(feature, sq, mask, out);
}